// DotPredictor_2010044695330
// MI455X (gfx1250) — compile-verified
//
#include <hip/hip_runtime.h>
#include <hip/hip_bf16.h>

typedef __attribute__((ext_vector_type(2))) float v2f;
typedef __attribute__((ext_vector_type(8))) float v8f;

// One wave (32 lanes) computes 16 edge dot-products via V_WMMA_F32_16X16X4_F32.
//
// Per K-chunk kb (kb = 0,4,...,60):
//   A[m][k]  = h[src[ebase+m]][kb+k]   (16x4)
//   B[k][n]  = h[dst[ebase+n]][kb+k]   (4x16)
//   D = A*B + C  accumulates; diagonal D[e][e] = dot(h[src[e]], h[dst[e]]) over kb.
//
// A layout (ISA 7.12.2, 32-bit A 16x4): lane m   -> K=0,1 in v[0],v[1]
//                                       lane m+16-> K=2,3
// B layout (row-striped like C/D):      lane n   -> K=0,1 in v[0],v[1]
//                                       lane n+16-> K=2,3
// => both A and B are a single float2 load per lane per chunk, base offset 2*(lane>>4).
__global__ __launch_bounds__(256) void edge_dot_wmma_kernel(
    const float* __restrict__ h,
    const int*   __restrict__ src,
    const int*   __restrict__ dst,
    float*       __restrict__ out,
    int n_edges)
{
    const int lane    = threadIdx.x & 31;
    const int waveId  = threadIdx.x >> 5;
    const int tile    = blockIdx.x * (blockDim.x >> 5) + waveId; // 16 edges per tile
    const int ebase   = tile * 16;
    if (ebase >= n_edges) return;              // wave-uniform exit: EXEC stays all-1s

    const int idx  = lane & 15;                // which edge of the tile this lane feeds
    const int half = lane >> 4;                // 0: K=0,1  1: K=2,3 within each 4-chunk

    int e = ebase + idx;
    if (e >= n_edges) e = n_edges - 1;         // clamp loads; keeps EXEC full for WMMA

    const int s = src[e];
    const int d = dst[e];
    const float* pu = h + (size_t)s * 64 + (half * 2);
    const float* pv = h + (size_t)d * 64 + (half * 2);

    v8f acc = {};                              // C = 0

    #pragma unroll
    for (int kb = 0; kb < 64; kb += 4) {
        v2f a = *(const v2f*)(pu + kb);
        v2f b = *(const v2f*)(pv + kb);
        // 8 args: (neg_a, A, neg_b, B, c_mod, C, reuse_a, reuse_b); NEG must be 0 for f32
        acc = __builtin_amdgcn_wmma_f32_16x16x4_f32(
            false, a, false, b, (short)0, acc, false, false);
    }

    // Diagonal extraction from 16x16 f32 C/D layout:
    //   m in 0..7  -> VGPR m,   lane m        (lanes 0..7)
    //   m in 8..15 -> VGPR m-8, lane m+16     (lanes 24..31)
    int m = -1;
    if (lane < 8)        m = lane;
    else if (lane >= 24) m = lane - 16;

    if (m >= 0) {
        const int r = m & 7;                   // which accumulator VGPR holds (m,m)
        float val = acc[0];
        #pragma unroll
        for (int i = 1; i < 8; ++i)
            if (r == i) val = acc[i];          // short v_cndmask select chain
        const int eo = ebase + m;
        if (eo < n_edges) out[eo] = val;
    }
}

extern "C" void kernel_launch(void* const* d_in, const int* in_sizes, int n_in,
                              void* d_out, int out_size, void* d_ws, size_t ws_size,
                              hipStream_t stream)
{
    const float* h   = (const float*)d_in[0];
    const int*   src = (const int*)d_in[1];
    const int*   dst = (const int*)d_in[2];
    float*       out = (float*)d_out;

    const int n_edges = in_sizes[1];           // 800000
    const int tiles   = (n_edges + 15) / 16;   // 16 edges per wave
    const int wavesPerBlock = 8;               // 256 threads
    const int blocks  = (tiles + wavesPerBlock - 1) / wavesPerBlock;

    edge_dot_wmma_kernel<<<blocks, 256, 0, stream>>>(h, src, dst, out, n_edges);
}